// Monomial_64716567216740
// MI455X (gfx1250) — compile-verified
//
#include <hip/hip_runtime.h>

typedef __attribute__((ext_vector_type(2))) float v2f;
typedef __attribute__((ext_vector_type(8))) float v8f;

namespace {
constexpr int    Bn         = 2;
constexpr int    Mn         = 50000;
constexpr int    FIN        = 128;
constexpr int    FOUT       = 128;
constexpr int    Kh         = 5;
constexpr int    NNZ        = 800000;
constexpr int    FB         = FIN * Bn;      // 256 (x0 row width)
constexpr int    ROWS       = Bn * Mn;       // 100000 GEMM rows
constexpr int    NODE_TILES = Mn / 16;       // 3125 (exact)
constexpr int    LDA        = 132;           // padded LDS stride: 132%64=4 -> conflict-free
constexpr int    GEMM_GRID  = 625;           // persistent blocks: 3125/625 = 5 tiles each
constexpr size_t XELEMS     = (size_t)Mn * FB;       // 12.8M floats per hop buffer
constexpr size_t YELEMS     = (size_t)ROWS * FOUT;   // 12.8M floats
}

__device__ __forceinline__ void atomic_add_f32(float* p, float v) {
    __hip_atomic_fetch_add(p, v, __ATOMIC_RELAXED, __HIP_MEMORY_SCOPE_AGENT);
}

// ---------------- zero fill (float / int) ----------------
__global__ __launch_bounds__(256) void zero_kernel(float* __restrict__ p, size_t n) {
    size_t i = (size_t)blockIdx.x * blockDim.x + threadIdx.x;
    if (i < n) p[i] = 0.0f;
}
__global__ __launch_bounds__(256) void zero_i32_kernel(int* __restrict__ p, size_t n) {
    size_t i = (size_t)blockIdx.x * blockDim.x + threadIdx.x;
    if (i < n) p[i] = 0;
}

// ---------------- build x0: X0[m][fin*2+b] = x[b][m][fin]; float2 writes ----------------
__global__ __launch_bounds__(256) void transpose_kernel(const float* __restrict__ x,
                                                        float* __restrict__ X0) {
    size_t t = (size_t)blockIdx.x * blockDim.x + threadIdx.x;   // 0 .. Mn*FIN-1
    if (t >= (size_t)Mn * FIN) return;
    int    fin = (int)(t & (FIN - 1));
    size_t m   = t >> 7;
    float2 v;
    v.x = x[m * FIN + fin];                       // b = 0 plane (coalesced)
    v.y = x[((size_t)Mn + m) * FIN + fin];        // b = 1 plane (coalesced)
    ((float2*)X0)[t] = v;                         // X0[m*256 + fin*2 + {0,1}]
}

// ---- permute weights into K-step-paired layout (see round-1 notes) -----------------------
__global__ __launch_bounds__(256) void rearrange_w_kernel(const float* __restrict__ W,
                                                          float* __restrict__ Wp) {
    int t = blockIdx.x * blockDim.x + threadIdx.x;   // 0 .. Kh*64*128-1 (float2 elements)
    if (t >= Kh * 64 * FOUT) return;
    int fout = t & (FOUT - 1);
    int idx  = (t >> 7) & 63;                        // s*2 + h
    int k    = t >> 13;
    int fin0 = (idx >> 1) * 4 + (idx & 1) * 2;
    float2 v;
    v.x = W[((size_t)fin0 * Kh + k) * FOUT + fout];
    v.y = W[((size_t)(fin0 + 1) * Kh + k) * FOUT + fout];
    ((float2*)Wp)[t] = v;
}

// ================= CSR build (once per launch) ===========================================
// pass 1: per-row edge histogram (native u32 atomics)
__global__ __launch_bounds__(256) void count_kernel(const int* __restrict__ rows,
                                                    int* __restrict__ cnt) {
    int e = blockIdx.x * blockDim.x + threadIdx.x;
    if (e < NNZ) atomicAdd(&cnt[rows[e]], 1);
}

// pass 2: exclusive scan of 50000 counters -> rowPtr[0..Mn]; single 1024-thread block
__global__ __launch_bounds__(1024) void scan_kernel(const int* __restrict__ cnt,
                                                    int* __restrict__ rowPtr) {
    __shared__ int buf[1024];
    __shared__ int carry;
    if (threadIdx.x == 0) carry = 0;
    __syncthreads();
    for (int base = 0; base < Mn; base += 1024) {
        int i = base + (int)threadIdx.x;
        int v = (i < Mn) ? cnt[i] : 0;
        buf[threadIdx.x] = v;
        __syncthreads();
        for (int off = 1; off < 1024; off <<= 1) {       // Hillis-Steele inclusive scan
            int add = (threadIdx.x >= (unsigned)off) ? buf[threadIdx.x - off] : 0;
            __syncthreads();
            buf[threadIdx.x] += add;
            __syncthreads();
        }
        int incl = buf[threadIdx.x];
        if (i < Mn) rowPtr[i] = carry + incl - v;        // exclusive
        __syncthreads();                                  // all reads of carry done
        if (threadIdx.x == 1023) carry += incl;           // incl == chunk total
        __syncthreads();
    }
    if (threadIdx.x == 0) rowPtr[Mn] = carry;
}

// pass 3: bin edges into CSR slots as packed (col, val_bits) 8-byte records
__global__ __launch_bounds__(256) void scatter_kernel(const int*   __restrict__ rows,
                                                      const int*   __restrict__ cols,
                                                      const float* __restrict__ vals,
                                                      const int*   __restrict__ rowPtr,
                                                      int*  __restrict__ cursor,
                                                      int2* __restrict__ edges) {
    int e = blockIdx.x * blockDim.x + threadIdx.x;
    if (e >= NNZ) return;
    int r   = rows[e];
    int pos = rowPtr[r] + atomicAdd(&cursor[r], 1);
    edges[pos] = make_int2(cols[e], __float_as_int(vals[e]));
}

// ---------------- gather SpMM hop: Xnext[r] = sum_{e in row r} val_e * Xcur[col_e] -------
// One wave32 per row; 8 accum regs/lane (row width 256); zero fp atomics; rows fully
// written (including empty rows) so no zero-fill pass is needed.
__global__ __launch_bounds__(256) void spmm_csr_kernel(const float* __restrict__ Xcur,
                                                       const int*   __restrict__ rowPtr,
                                                       const int2*  __restrict__ edges,
                                                       float* __restrict__ Xnext) {
    int r    = (blockIdx.x * blockDim.x + threadIdx.x) >> 5;   // 50000/8 rows per block: exact
    int lane = threadIdx.x & 31;
    if (r >= Mn) return;
    const int beg = rowPtr[r], end = rowPtr[r + 1];
    float acc[FB / 32] = {};
    for (int j = beg; j < end; ++j) {
        int2   e    = edges[j];                    // b64 broadcast load (uniform in wave)
        float  val  = __int_as_float(e.y);
        size_t cb   = (size_t)e.x * FB;
#pragma unroll
        for (int q = 0; q < FB / 32; ++q)          // coalesced 128B gathers, L2-resident
            acc[q] = fmaf(val, Xcur[cb + lane + q * 32], acc[q]);
    }
    size_t rb = (size_t)r * FB;
#pragma unroll
    for (int q = 0; q < FB / 32; ++q)
        Xnext[rb + lane + q * 32] = acc[q];
}

// ---------------- GEMM hop (persistent): y += X_k @ W_k via V_WMMA_F32_16X16X4_F32 -------
__global__ __launch_bounds__(512) void gemm_hop_kernel(const float* __restrict__ Xc,
                                                       const float* __restrict__ Wpk,
                                                       float* __restrict__ y,
                                                       float* __restrict__ pooledSum,
                                                       int do_pool) {
    __shared__ float Atile[Bn][16 * LDA];
    const int tid  = threadIdx.x;
    const int wave = tid >> 5;                 // 0..15
    const int lane = tid & 31;
    const int half = lane >> 4;
    const int l16  = lane & 15;
    const int wb   = wave >> 3;                // batch owned by this wave
    const int fout = (wave & 7) * 16 + l16;

    v2f breg[32];                              // B slab: 64 VGPRs/lane, loaded once
    const v2f* W2 = (const v2f*)Wpk;
#pragma unroll
    for (int s = 0; s < 32; ++s)
        breg[s] = W2[(s * 2 + half) * FOUT + fout];

    for (int t = blockIdx.x; t < NODE_TILES; t += gridDim.x) {
        const int node0 = t * 16;
        for (int i = tid; i < 16 * FIN; i += 512) {
            int    fin = i & (FIN - 1);
            int    m   = i >> 7;
            float2 v   = ((const float2*)Xc)[(size_t)(node0 + m) * FIN + fin];
            Atile[0][m * LDA + fin] = v.x;
            Atile[1][m * LDA + fin] = v.y;
        }
        __syncthreads();

        const float* At = Atile[wb];
        v8f acc = {};
#pragma unroll
        for (int s = 0; s < 32; ++s) {
            const int kk = s * 4 + half * 2;
            v2f a;
            a.x = At[l16 * LDA + kk];
            a.y = At[l16 * LDA + kk + 1];
            acc = __builtin_amdgcn_wmma_f32_16x16x4_f32(
                      false, a, false, breg[s], (short)0, acc, false, false);
        }

        const size_t rowbase = (size_t)wb * Mn + node0;
        float colsum = 0.0f;
#pragma unroll
        for (int r = 0; r < 8; ++r) {
            size_t idx = (rowbase + r + half * 8) * FOUT + fout;
            float  v   = y[idx] + acc[r];
            y[idx]     = v;
            colsum    += v;
        }
        if (do_pool) atomic_add_f32(&pooledSum[wb * FOUT + fout], colsum);
        __syncthreads();
    }
}

// ---------------- SE gate ----------------------------------------------------------------
__global__ __launch_bounds__(128) void gate_kernel(const float* __restrict__ pooledSum,
                                                   const float* __restrict__ dw,
                                                   const float* __restrict__ db,
                                                   const float* __restrict__ uw,
                                                   const float* __restrict__ ub,
                                                   float* __restrict__ gate) {
    __shared__ float p[FOUT];
    __shared__ float h[FOUT / 4];
    const int b = blockIdx.x, f = threadIdx.x;
    p[f] = pooledSum[b * FOUT + f] * (1.0f / (float)Mn);
    __syncthreads();
    if (f < FOUT / 4) {
        float s = db[f];
        for (int i = 0; i < FOUT; ++i) s += p[i] * dw[i * (FOUT / 4) + f];
        h[f] = s / (1.0f + expf(-s));                         // silu
    }
    __syncthreads();
    float s = ub[f];
    for (int i = 0; i < FOUT / 4; ++i) s += h[i] * uw[i * FOUT + f];
    gate[b * FOUT + f] = 1.0f / (1.0f + expf(-s));            // sigmoid
}

// ---------------- out *= gate, float4-vectorized -----------------------------------------
__global__ __launch_bounds__(256) void scale_kernel(float* __restrict__ y,
                                                    const float* __restrict__ gate) {
    size_t i = (size_t)blockIdx.x * blockDim.x + threadIdx.x;  // float4 index
    if (i >= YELEMS / 4) return;
    int    f0 = (int)((i * 4) & (FOUT - 1));
    int    b  = (int)(i / ((size_t)Mn * FOUT / 4));
    float4 v  = ((float4*)y)[i];
    v.x *= gate[b * FOUT + f0 + 0];
    v.y *= gate[b * FOUT + f0 + 1];
    v.z *= gate[b * FOUT + f0 + 2];
    v.w *= gate[b * FOUT + f0 + 3];
    ((float4*)y)[i] = v;
}

extern "C" void kernel_launch(void* const* d_in, const int* in_sizes, int n_in,
                              void* d_out, int out_size, void* d_ws, size_t ws_size,
                              hipStream_t stream) {
    const float* x      = (const float*)d_in[0];
    const float* vals   = (const float*)d_in[1];
    const float* kernW  = (const float*)d_in[2];
    const float* down_w = (const float*)d_in[3];
    const float* down_b = (const float*)d_in[4];
    const float* up_w   = (const float*)d_in[5];
    const float* up_b   = (const float*)d_in[6];
    const int*   rowsI  = (const int*)d_in[7];
    const int*   colsI  = (const int*)d_in[8];
    float*       y      = (float*)d_out;

    // Workspace (floats): X0 | X1 | Wp | pooledSum | gate | rowPtr | cursor | edges
    float* ws        = (float*)d_ws;
    float* X0        = ws;
    float* X1        = ws + XELEMS;
    float* Wp        = ws + 2 * XELEMS;
    float* pooledSum = Wp + (size_t)Kh * FIN * FOUT;
    float* gate      = pooledSum + Bn * FOUT;
    int*   rowPtr    = (int*)(gate + Bn * FOUT);         // Mn+1 (padded to Mn+2)
    int*   cursor    = rowPtr + (Mn + 2);                // Mn   (padded to Mn+2)
    int2*  edges     = (int2*)(cursor + (Mn + 2));       // NNZ records, 8B-aligned

    const int TPB = 256;
    const int zgY = (int)((YELEMS + TPB - 1) / TPB);

    zero_kernel<<<zgY, TPB, 0, stream>>>(y, YELEMS);
    zero_kernel<<<2, TPB, 0, stream>>>(pooledSum, (size_t)(2 * Bn * FOUT));
    transpose_kernel<<<(int)(((size_t)Mn * FIN + TPB - 1) / TPB), TPB, 0, stream>>>(x, X0);
    rearrange_w_kernel<<<(Kh * 64 * FOUT + TPB - 1) / TPB, TPB, 0, stream>>>(kernW, Wp);

    // Build CSR once: histogram -> scan -> bin (cursor doubles as the histogram buffer)
    zero_i32_kernel<<<(Mn + 2 + TPB - 1) / TPB, TPB, 0, stream>>>(cursor, Mn + 2);
    count_kernel<<<(NNZ + TPB - 1) / TPB, TPB, 0, stream>>>(rowsI, cursor);
    scan_kernel<<<1, 1024, 0, stream>>>(cursor, rowPtr);
    zero_i32_kernel<<<(Mn + 2 + TPB - 1) / TPB, TPB, 0, stream>>>(cursor, Mn + 2);
    scatter_kernel<<<(NNZ + TPB - 1) / TPB, TPB, 0, stream>>>(rowsI, colsI, vals,
                                                              rowPtr, cursor, edges);

    float* Xcur = X0;
    float* Xnext = X1;
    for (int k = 0; k < Kh; ++k) {
        gemm_hop_kernel<<<GEMM_GRID, 512, 0, stream>>>(
            Xcur, Wp + (size_t)k * FIN * FOUT, y, pooledSum, (k == Kh - 1) ? 1 : 0);
        if (k < Kh - 1) {
            spmm_csr_kernel<<<(Mn * 32) / TPB, TPB, 0, stream>>>(Xcur, rowPtr, edges, Xnext);
            float* t = Xcur; Xcur = Xnext; Xnext = t;
        }
    }

    gate_kernel<<<Bn, 128, 0, stream>>>(pooledSum, down_w, down_b, up_w, up_b, gate);
    scale_kernel<<<(int)((YELEMS / 4 + TPB - 1) / TPB), TPB, 0, stream>>>(y, gate);
}